// Loss_5729486373283
// MI455X (gfx1250) — compile-verified
//
#include <hip/hip_runtime.h>
#include <math.h>

// ---- problem constants (from reference) ----
#define IMG_SIZE_F 640.0f
#define KPT 17          // keypoints
#define NT 20           // targets per batch
#define TGT_STRIDE 56   // 4 box + 1 conf + 34 kpts + 17 vis
#define AN 3            // anchors per level
#define FEAT 39         // channels per anchor (117/3)
#define NBATCH 8

typedef __attribute__((ext_vector_type(2))) float v2f;
typedef __attribute__((ext_vector_type(8))) float v8f;
typedef __attribute__((ext_vector_type(4))) unsigned int v4u;
typedef __attribute__((ext_vector_type(4))) int v4i;
typedef __attribute__((ext_vector_type(8))) int v8i;

#if __has_builtin(__builtin_amdgcn_tensor_load_to_lds) && \
    __has_builtin(__builtin_amdgcn_s_wait_tensorcnt)
#define USE_TDM 1
#else
#define USE_TDM 0
#endif

// Zero the 16 per-level accumulators (4 levels x {sum_iou, sum_bce, sum_l1, sum_tm})
__global__ void zero_acc_kernel(float* acc) {
    if (threadIdx.x < 16) acc[threadIdx.x] = 0.0f;
}

// One thread per (anchor, h, w); blockIdx.y = batch. Block = 256 threads = 8 waves.
__global__ void level_loss_kernel(const float* __restrict__ pred,
                                  const float* __restrict__ targets,
                                  const float* __restrict__ anchors,  // 6 floats for this level
                                  float* __restrict__ acc,            // 4 floats for this level
                                  int H)
{
    const int W  = H;
    const int HW = H * W;
    const int total = AN * HW;
    const int b   = blockIdx.y;
    const int tid = threadIdx.x;
    const int idx = blockIdx.x * blockDim.x + tid;

    __shared__ float sT[NT * TGT_STRIDE]; // raw target rows for this batch
    __shared__ float sTC[NT][4];          // round-tripped corner boxes
    __shared__ float sArea[NT];           // target areas (from round-tripped corners)
    __shared__ float sAnch[AN * 2];
    __shared__ float sP[8][4];            // per-wave partials: [wave][accumulator]

#if USE_TDM
    // Stage the 20 target rows (1120 f32 = 4480 B) into LDS via the Tensor Data
    // Mover: 1-D tile descriptor, issued once by wave 0; TENSORcnt tracks it and
    // __syncthreads() publishes the LDS contents to the other 7 waves.
    if (tid < 32) {
        const unsigned lds_off = (unsigned)(size_t)(void*)&sT[0];   // flat->LDS: low 32 bits
        const unsigned long long ga =
            (unsigned long long)(size_t)(const void*)(targets + (size_t)b * (NT * TGT_STRIDE));
        const unsigned nelem = NT * TGT_STRIDE;                     // 1120 elements of 4 B

        v4u g0;
        g0.x = 1u;                                    // count=1 (valid descriptor), user mode
        g0.y = lds_off;                               // lds_addr (bytes)
        g0.z = (unsigned)(ga & 0xFFFFFFFFull);        // global_addr[31:0]
        g0.w = (unsigned)((ga >> 32) & 0x01FFFFFFull) // global_addr[56:32]
             | 0x80000000u;                           // type=2 ("image") in bits 127:126

        v8i g1;
        g1[0] = (int)(2u << 16);                      // data_size=2 -> 4-byte elements
        g1[1] = (int)((nelem & 0xFFFFu) << 16);       // tensor_dim0[15:0]   (bits 63:48)
        g1[2] = (int)(1u << 16);                      // tensor_dim1 = 1     (bits 95:80)
        g1[3] = (int)((nelem & 0xFFFFu) << 16);       // tile_dim0 = 1120    (bits 127:112)
        g1[4] = 0;                                    // tile_dim1=0, tile_dim2=0 (unused)
        g1[5] = (int)nelem;                           // tensor_dim0_stride[31:0]
        g1[6] = 0;                                    // stride[47:32]=0, dim1_stride lo=0
        g1[7] = 0;                                    // dim1_stride hi=0

        v4i g2 = {0, 0, 0, 0};                        // <=2-D tensor: groups 2/3 unused
        v4i g3 = {0, 0, 0, 0};
#if __clang_major__ >= 23
        v8i g4 = {0, 0, 0, 0, 0, 0, 0, 0};
        __builtin_amdgcn_tensor_load_to_lds(g0, g1, g2, g3, g4, 0);
#else
        __builtin_amdgcn_tensor_load_to_lds(g0, g1, g2, g3, 0);
#endif
        __builtin_amdgcn_s_wait_tensorcnt(0);         // s_wait_tensorcnt 0
    }
    if (tid < AN * 2) sAnch[tid] = anchors[tid];
    __syncthreads();
#else
    // Fallback: cooperative coalesced staging.
    for (int i = tid; i < NT * TGT_STRIDE; i += blockDim.x)
        sT[i] = targets[(size_t)b * (NT * TGT_STRIDE) + i];
    if (tid < AN * 2) sAnch[tid] = anchors[tid];
    __syncthreads();
#endif

    // Precompute corner round-trip (matches reference fp ordering) + areas.
    if (tid < NT) {
        const float* t = &sT[tid * TGT_STRIDE];
        float cx = (t[0] + t[2]) * 0.5f;
        float cy = (t[1] + t[3]) * 0.5f;
        float w  = t[2] - t[0];
        float h  = t[3] - t[1];
        float x1 = cx - w * 0.5f, y1 = cy - h * 0.5f;
        float x2 = cx + w * 0.5f, y2 = cy + h * 0.5f;
        sTC[tid][0] = x1; sTC[tid][1] = y1; sTC[tid][2] = x2; sTC[tid][3] = y2;
        sArea[tid] = (x2 - x1) * (y2 - y1);
    }
    __syncthreads();

    float sum_iou = 0.0f, sum_bce = 0.0f, sum_l1 = 0.0f, sum_tm = 0.0f;

    if (idx < total) {
        const int a   = idx / HW;
        const int rem = idx - a * HW;
        const int h   = rem / W;
        const int w   = rem - h * W;

        const size_t s    = (size_t)HW;                       // channel stride
        const size_t base = ((size_t)b * 117 + (size_t)a * FEAT) * s
                          + (size_t)h * W + (size_t)w;

        const float px    = pred[base];
        const float py    = pred[base + s];
        const float pw    = pred[base + 2 * s];
        const float ph    = pred[base + 3 * s];
        const float pconf = pred[base + 4 * s];

        const float scale = IMG_SIZE_F / (float)W;
        const float bx = (1.0f / (1.0f + expf(-px)) + (float)w) * scale;
        const float by = (1.0f / (1.0f + expf(-py)) + (float)h) * scale;
        const float bw = expf(pw) * sAnch[a * 2 + 0];
        const float bh = expf(ph) * sAnch[a * 2 + 1];

        const float pc0 = bx - bw * 0.5f, pc1 = by - bh * 0.5f;
        const float pc2 = bx + bw * 0.5f, pc3 = by + bh * 0.5f;
        const float a1  = (pc2 - pc0) * (pc3 - pc1);

        // Match: max IoU over 20 targets (first-occurrence argmax like jnp).
        float best = -1.0f; int bestIdx = 0;
        for (int t = 0; t < NT; ++t) {
            float xi1 = fmaxf(pc0, sTC[t][0]);
            float yi1 = fmaxf(pc1, sTC[t][1]);
            float xi2 = fminf(pc2, sTC[t][2]);
            float yi2 = fminf(pc3, sTC[t][3]);
            float inter = fmaxf(xi2 - xi1, 0.0f) * fmaxf(yi2 - yi1, 0.0f);
            float iou = inter / (a1 + sArea[t] - inter + 1e-9f);
            if (iou > best) { best = iou; bestIdx = t; }
        }
        const bool mask = best > 0.5f;

        // loss_box term: IoU(pred corners, matched-or-zero box corners)
        float t0 = 0.f, t1 = 0.f, t2 = 0.f, t3 = 0.f;
        if (mask) {
            t0 = sTC[bestIdx][0]; t1 = sTC[bestIdx][1];
            t2 = sTC[bestIdx][2]; t3 = sTC[bestIdx][3];
        }
        {
            float xi1 = fmaxf(pc0, t0), yi1 = fmaxf(pc1, t1);
            float xi2 = fminf(pc2, t2), yi2 = fminf(pc3, t3);
            float inter = fmaxf(xi2 - xi1, 0.0f) * fmaxf(yi2 - yi1, 0.0f);
            float a2 = (t2 - t0) * (t3 - t1);
            sum_iou = inter / (a1 + a2 - inter + 1e-9f);
        }

        // loss_conf term: BCE with logits, target = mask
        {
            const float tc = mask ? 1.0f : 0.0f;
            sum_bce = fmaxf(pconf, 0.0f) - pconf * tc + log1pf(expf(-fabsf(pconf)));
        }

        // loss_kpts: only matched cells touch the 34 keypoint channels
        if (mask) {
            const float* trow = &sT[bestIdx * TGT_STRIDE];
            for (int k = 0; k < KPT; ++k) {
                if (trow[39 + k] > 0.0f) {
                    float kx = pred[base + (size_t)(5 + 2 * k) * s];
                    float ky = pred[base + (size_t)(6 + 2 * k) * s];
                    sum_l1 += fabsf(kx - trow[5 + 2 * k]) + fabsf(ky - trow[6 + 2 * k]);
                    sum_tm += 2.0f;
                }
            }
        }
    }

    // ---- wave32 butterfly reduction ----
    for (int off = 16; off > 0; off >>= 1) {
        sum_iou += __shfl_xor(sum_iou, off, 32);
        sum_bce += __shfl_xor(sum_bce, off, 32);
        sum_l1  += __shfl_xor(sum_l1,  off, 32);
        sum_tm  += __shfl_xor(sum_tm,  off, 32);
    }
    const int wave = tid >> 5;
    if ((tid & 31) == 0) {
        sP[wave][0] = sum_iou; sP[wave][1] = sum_bce;
        sP[wave][2] = sum_l1;  sP[wave][3] = sum_tm;
    }
    __syncthreads();

    // ---- cross-wave combine on the matrix pipe ----
    // 8 wave-partials x 4 accumulators packed into a 16x4 f32 A-tile
    // (row m=2q+r holds accumulator q's partials for waves r*4..r*4+3),
    // multiplied by all-ones B: D row-sums give all 4 block totals.
    if (tid < 32) {
        const int lane = tid;
        const int m = lane & 15;
        v2f A = {0.0f, 0.0f};
        if (m < 8) {
            const int q = m >> 1;
            const int r = m & 1;
            const int kbase = (lane < 16) ? 0 : 2;   // lanes 0-15: K=0,1; lanes 16-31: K=2,3
            A.x = sP[r * 4 + kbase + 0][q];
            A.y = sP[r * 4 + kbase + 1][q];
        }
        v2f Bm = {1.0f, 1.0f};
        v8f C = {};
#if __has_builtin(__builtin_amdgcn_wmma_f32_16x16x4_f32)
        v8f D = __builtin_amdgcn_wmma_f32_16x16x4_f32(
            /*neg_a=*/false, A, /*neg_b=*/false, Bm,
            /*c_mod=*/(short)0, C, /*reuse_a=*/false, /*reuse_b=*/false);
        if (lane == 0) {
            atomicAdd(&acc[0], D[0] + D[1]);   // sum_iou
            atomicAdd(&acc[1], D[2] + D[3]);   // sum_bce
            atomicAdd(&acc[2], D[4] + D[5]);   // sum_l1
            atomicAdd(&acc[3], D[6] + D[7]);   // sum_tm
        }
#else
        (void)Bm; (void)C;
        if (lane == 0) {
            for (int q = 0; q < 4; ++q) {
                float v = 0.0f;
                for (int wv = 0; wv < 8; ++wv) v += sP[wv][q];
                atomicAdd(&acc[q], v);
            }
        }
#endif
    }
}

// Combine the 16 accumulators into the scalar loss.
__global__ void finalize_kernel(const float* __restrict__ acc, float* __restrict__ out) {
    const float bal[4] = {4.0f, 1.0f, 0.4f, 0.1f};
    const int   Hs[4]  = {160, 80, 40, 20};
    float total = 0.0f;
    for (int l = 0; l < 4; ++l) {
        const float cnt = (float)NBATCH * AN * Hs[l] * Hs[l];  // B*H*W*A
        const float sz  = cnt * (2.0f * KPT);                  // kpred.size
        const float s_iou = acc[l * 4 + 0];
        const float s_bce = acc[l * 4 + 1];
        const float s_l1  = acc[l * 4 + 2];
        const float msum  = acc[l * 4 + 3];
        const float loss_box  = 1.0f - s_iou / cnt;
        const float loss_conf = s_bce / cnt;
        const float loss_kpts = (msum > 0.0f) ? (s_l1 / sz) / msum : 0.0f;
        total += bal[l] * (loss_box + loss_conf + loss_kpts);
    }
    out[0] = total / (float)NBATCH;
}

extern "C" void kernel_launch(void* const* d_in, const int* in_sizes, int n_in,
                              void* d_out, int out_size, void* d_ws, size_t ws_size,
                              hipStream_t stream) {
    (void)in_sizes; (void)n_in; (void)out_size; (void)ws_size;

    const float* preds[4] = { (const float*)d_in[0], (const float*)d_in[1],
                              (const float*)d_in[2], (const float*)d_in[3] };
    const float* targets = (const float*)d_in[4];
    const float* anchors = (const float*)d_in[5];
    float* acc = (float*)d_ws;        // 16 floats of scratch

    zero_acc_kernel<<<1, 32, 0, stream>>>(acc);

    const int Hs[4] = {160, 80, 40, 20};
    for (int l = 0; l < 4; ++l) {
        const int total = AN * Hs[l] * Hs[l];
        dim3 grid((total + 255) / 256, NBATCH);
        level_loss_kernel<<<grid, 256, 0, stream>>>(
            preds[l], targets, anchors + l * 6, acc + l * 4, Hs[l]);
    }

    finalize_kernel<<<1, 1, 0, stream>>>(acc, (float*)d_out);
}